// QuantumSuperpositionAttention_31791347925713
// MI455X (gfx1250) — compile-verified
//
#include <hip/hip_runtime.h>
#include <hip/hip_bf16.h>

// ---------------------------------------------------------------------------
// QuantumSuperpositionAttention on gfx1250 (MI455X, wave32, WMMA).
//
// Key optimization: beta = alpha * exp(i*phase) => |beta|^2 == |alpha|^2, so
// the phase-modulation MLP is dead code and is eliminated entirely.
//
// Pipeline:
//   1) cgemm_xw    : complex X @ {Wq,Wk,Wv}  -> Re/Im planes (WMMA f32 16x16x4)
//   2) scores_head : p = scale^2 * |conj(Q_h) . K_h|^2, fused row-sum atomics
//   3) topk_route  : exact top-102 per row, eps-renorm, softmax(k*p), weighted
//                    complex gather of V                (LDS / ds ops)
//   4) cgemm_ow    : complex O @ Wo -> interleaved complex64 output
// ---------------------------------------------------------------------------

#define N_TOK 1024
#define DIMC  512
#define NHEAD 8
#define DHEAD 64
#define KSEL  102           // max(1, int(0.1 * 1024))
#define EPSV  1e-8f

typedef __attribute__((ext_vector_type(2))) float v2f;
typedef __attribute__((ext_vector_type(8))) float v8f;

__device__ __forceinline__ v8f wmma_f32(v2f a, v2f b, v8f c) {
  // V_WMMA_F32_16X16X4_F32 : D = A(16x4) * B(4x16) + C(16x16)
  return __builtin_amdgcn_wmma_f32_16x16x4_f32(false, a, false, b, (short)0, c,
                                               false, false);
}

// ---------------------------------------------------------------------------
// Complex GEMM: X (interleaved complex, M x K) @ W (interleaved complex, K x K)
// -> separate Re / Im planes. One wave computes one 16x16 output tile.
// grid = (DIMC/16, N_TOK/16), block = 32.
// ---------------------------------------------------------------------------
__global__ void cgemm_xw(const float* __restrict__ X, const float* __restrict__ W,
                         float* __restrict__ outR, float* __restrict__ outI) {
  const int lane  = threadIdx.x;          // 0..31
  const int nbase = blockIdx.x * 16;      // output column tile
  const int mbase = blockIdx.y * 16;      // output row tile
  const int rc = lane & 15;               // row (A) / col (B,D) within tile
  const int kq = (lane >> 4) * 2;         // K sub-offset per half-wave

  v8f cr = {0.f,0.f,0.f,0.f,0.f,0.f,0.f,0.f};
  v8f ci = {0.f,0.f,0.f,0.f,0.f,0.f,0.f,0.f};

  for (int k = 0; k < DIMC; k += 4) {
    const float* xp = X + 2 * ((size_t)(mbase + rc) * DIMC + k + kq);
    v2f ar = { xp[0], xp[2] };            // Re x[row, k+kq .. k+kq+1]
    v2f ai = { xp[1], xp[3] };            // Im
    const float* w0 = W + 2 * ((size_t)(k + kq)     * DIMC + nbase + rc);
    const float* w1 = W + 2 * ((size_t)(k + kq + 1) * DIMC + nbase + rc);
    v2f br = { w0[0], w1[0] };
    v2f bi = { w0[1], w1[1] };
    v2f nai = { -ai.x, -ai.y };
    cr = wmma_f32(ar,  br, cr);           // + ArBr
    cr = wmma_f32(nai, bi, cr);           // - AiBi
    ci = wmma_f32(ar,  bi, ci);           // + ArBi
    ci = wmma_f32(ai,  br, ci);           // + AiBr
  }

  const int mhi = (lane >> 4) * 8;
#pragma unroll
  for (int v = 0; v < 8; ++v) {
    const size_t o = (size_t)(mbase + v + mhi) * DIMC + nbase + rc;
    outR[o] = cr[v];
    outI[o] = ci[v];
  }
}

// ---------------------------------------------------------------------------
// Per-head scores: p[n,s] = scale^2 * |sum_d conj(Q[n,d]) K[s,d]|^2
// Re = Qr.Kr + Qi.Ki ; Im = Qr.Ki - Qi.Kr.  Fused row-sum via shfl + atomics.
// grid = (N/16 s-tiles, N/16 n-tiles), block = 32.
// ---------------------------------------------------------------------------
__global__ void scores_head(const float* __restrict__ Qr, const float* __restrict__ Qi,
                            const float* __restrict__ Kr, const float* __restrict__ Ki,
                            float* __restrict__ attn, float* __restrict__ rowsum,
                            int h) {
  const int lane  = threadIdx.x;
  const int sbase = blockIdx.x * 16;
  const int nbase = blockIdx.y * 16;
  const int rc = lane & 15;
  const int kq = (lane >> 4) * 2;
  const int hoff = h * DHEAD;
  const float scale = 0.125f;             // DH^-0.5

  v8f cre = {0.f,0.f,0.f,0.f,0.f,0.f,0.f,0.f};
  v8f cim = {0.f,0.f,0.f,0.f,0.f,0.f,0.f,0.f};

  for (int d = 0; d < DHEAD; d += 4) {
    const float* qrp = Qr + (size_t)(nbase + rc) * DIMC + hoff + d + kq;
    const float* qip = Qi + (size_t)(nbase + rc) * DIMC + hoff + d + kq;
    v2f aqr = { qrp[0], qrp[1] };
    v2f aqi = { qip[0], qip[1] };
    const float* krp = Kr + (size_t)(sbase + rc) * DIMC + hoff + d + kq;
    const float* kip = Ki + (size_t)(sbase + rc) * DIMC + hoff + d + kq;
    v2f bkr = { krp[0], krp[1] };         // B[d, s] = K[s, d]  (transposed feed)
    v2f bki = { kip[0], kip[1] };
    v2f nqi = { -aqi.x, -aqi.y };
    cre = wmma_f32(aqr, bkr, cre);        // + QrKr
    cre = wmma_f32(aqi, bki, cre);        // + QiKi
    cim = wmma_f32(aqr, bki, cim);        // + QrKi
    cim = wmma_f32(nqi, bkr, cim);        // - QiKr
  }

  const int mhi = (lane >> 4) * 8;
#pragma unroll
  for (int v = 0; v < 8; ++v) {
    float re = cre[v] * scale;
    float im = cim[v] * scale;
    float p  = re * re + im * im;         // == |beta|^2 (phase MLP is dead)
    const int n = nbase + v + mhi;
    attn[(size_t)n * N_TOK + sbase + rc] = p;
    // row-sum over the 16 columns held by this 16-lane group
    float s = p;
#pragma unroll
    for (int off = 8; off >= 1; off >>= 1) s += __shfl_xor(s, off, 16);
    if (rc == 0) atomicAdd(&rowsum[n], s);
  }
}

// ---------------------------------------------------------------------------
// Exact top-k (k=102) routing per row + weighted complex gather of V.
// grid = N_TOK blocks of 256 threads; one block per (head-row).
// ---------------------------------------------------------------------------
__global__ void topk_route(const float* __restrict__ attn, const float* __restrict__ rowsum,
                           const float* __restrict__ Vr, const float* __restrict__ Vi,
                           float* __restrict__ Or, float* __restrict__ Oi, int h) {
  __shared__ float vals[N_TOK];
  __shared__ float red [256];
  __shared__ float red2[256];
  __shared__ int   redi[256];
  __shared__ float selv[KSEL];
  __shared__ int   seli[KSEL];
  __shared__ float wts [KSEL];

  const int n   = blockIdx.x;
  const int tid = threadIdx.x;

  const float inv = 1.0f / (rowsum[n] + EPSV);
  for (int i = tid; i < N_TOK; i += 256)
    vals[i] = attn[(size_t)n * N_TOK + i] * inv;
  __syncthreads();

  // iterative argmax; tie -> lower index (matches lax.top_k ordering)
  for (int it = 0; it < KSEL; ++it) {
    float bm = -1e30f; int bi = 0x7fffffff;
    for (int i = tid; i < N_TOK; i += 256) {
      float v = vals[i];
      if (v > bm) { bm = v; bi = i; }
    }
    red[tid] = bm; redi[tid] = bi;
    __syncthreads();
    for (int off = 128; off > 0; off >>= 1) {
      if (tid < off) {
        float ov = red[tid + off]; int oi = redi[tid + off];
        if (ov > red[tid] || (ov == red[tid] && oi < redi[tid])) {
          red[tid] = ov; redi[tid] = oi;
        }
      }
      __syncthreads();
    }
    if (tid == 0) {
      selv[it] = red[0]; seli[it] = redi[0];
      vals[redi[0]] = -1e30f;               // remove from candidates
    }
    __syncthreads();
  }

  // renormalize top-k (+eps) and softmax(topk_p * k); selv is descending so
  // selv[0] is the softmax max.
  if (tid == 0) {
    float s = 0.f;
    for (int j = 0; j < KSEL; ++j) s += selv[j];
    const float sc = (float)KSEL / (s + EPSV);
    const float t0 = selv[0] * sc;
    float es = 0.f;
    for (int j = 0; j < KSEL; ++j) {
      float e = __expf(selv[j] * sc - t0);
      wts[j] = e; es += e;
    }
    const float ie = 1.0f / es;
    for (int j = 0; j < KSEL; ++j) wts[j] *= ie;
  }
  __syncthreads();

  // out[n, h*64 + d] = sum_j w_j * V[idx_j, h*64 + d]   (real w, complex V)
  const int d     = tid & 63;
  const int slice = tid >> 6;               // 4 slices over j
  float ar = 0.f, ai = 0.f;
  for (int j = slice; j < KSEL; j += 4) {
    const float w = wts[j];
    const size_t off = (size_t)seli[j] * DIMC + h * DHEAD + d;
    ar += w * Vr[off];
    ai += w * Vi[off];
  }
  red[tid] = ar; red2[tid] = ai;
  __syncthreads();
  if (slice == 0) {
    float sr = red [tid] + red [tid + 64] + red [tid + 128] + red [tid + 192];
    float si = red2[tid] + red2[tid + 64] + red2[tid + 128] + red2[tid + 192];
    const size_t o = (size_t)n * DIMC + h * DHEAD + d;
    Or[o] = sr; Oi[o] = si;
  }
}

// ---------------------------------------------------------------------------
// Final complex GEMM: (Or + i*Oi) @ Wo (interleaved) -> interleaved complex64.
// grid = (DIMC/16, N_TOK/16), block = 32.
// ---------------------------------------------------------------------------
__global__ void cgemm_ow(const float* __restrict__ Ar, const float* __restrict__ Ai,
                         const float* __restrict__ W, float* __restrict__ out) {
  const int lane  = threadIdx.x;
  const int nbase = blockIdx.x * 16;
  const int mbase = blockIdx.y * 16;
  const int rc = lane & 15;
  const int kq = (lane >> 4) * 2;

  v8f cr = {0.f,0.f,0.f,0.f,0.f,0.f,0.f,0.f};
  v8f ci = {0.f,0.f,0.f,0.f,0.f,0.f,0.f,0.f};

  for (int k = 0; k < DIMC; k += 4) {
    const float* arp = Ar + (size_t)(mbase + rc) * DIMC + k + kq;
    const float* aip = Ai + (size_t)(mbase + rc) * DIMC + k + kq;
    v2f ar = { arp[0], arp[1] };
    v2f ai = { aip[0], aip[1] };
    const float* w0 = W + 2 * ((size_t)(k + kq)     * DIMC + nbase + rc);
    const float* w1 = W + 2 * ((size_t)(k + kq + 1) * DIMC + nbase + rc);
    v2f br = { w0[0], w1[0] };
    v2f bi = { w0[1], w1[1] };
    v2f nai = { -ai.x, -ai.y };
    cr = wmma_f32(ar,  br, cr);
    cr = wmma_f32(nai, bi, cr);
    ci = wmma_f32(ar,  bi, ci);
    ci = wmma_f32(ai,  br, ci);
  }

  const int mhi = (lane >> 4) * 8;
#pragma unroll
  for (int v = 0; v < 8; ++v) {
    const size_t o = 2 * ((size_t)(mbase + v + mhi) * DIMC + nbase + rc);
    out[o]     = cr[v];
    out[o + 1] = ci[v];
  }
}

// ---------------------------------------------------------------------------
extern "C" void kernel_launch(void* const* d_in, const int* in_sizes, int n_in,
                              void* d_out, int out_size, void* d_ws, size_t ws_size,
                              hipStream_t stream) {
  (void)in_sizes; (void)n_in; (void)out_size; (void)ws_size;

  const float* x  = (const float*)d_in[0];   // complex64 interleaved (N, DIM)
  const float* Wq = (const float*)d_in[1];   // complex64 interleaved (DIM, DIM)
  const float* Wk = (const float*)d_in[2];
  const float* Wv = (const float*)d_in[3];
  const float* Wo = (const float*)d_in[4];
  // d_in[5..8] = phase-MLP weights: provably dead (|alpha*e^{i phi}|^2 == |alpha|^2)

  const size_t NS = (size_t)N_TOK * DIMC;    // 524288 floats per plane
  float* ws = (float*)d_ws;
  float* Qr = ws + 0 * NS;  float* Qi = ws + 1 * NS;
  float* Kr = ws + 2 * NS;  float* Ki = ws + 3 * NS;
  float* Vr = ws + 4 * NS;  float* Vi = ws + 5 * NS;
  float* Or = ws + 6 * NS;  float* Oi = ws + 7 * NS;
  float* attn   = ws + 8 * NS;               // N*N = 2*NS floats
  float* rowsum = ws + 10 * NS;              // N floats

  dim3 gblk(32);
  dim3 gemm_grid(DIMC / 16, N_TOK / 16);     // (32, 64)

  cgemm_xw<<<gemm_grid, gblk, 0, stream>>>(x, Wq, Qr, Qi);
  cgemm_xw<<<gemm_grid, gblk, 0, stream>>>(x, Wk, Kr, Ki);
  cgemm_xw<<<gemm_grid, gblk, 0, stream>>>(x, Wv, Vr, Vi);

  dim3 score_grid(N_TOK / 16, N_TOK / 16);   // (64, 64)
  for (int h = 0; h < NHEAD; ++h) {
    hipMemsetAsync(rowsum, 0, N_TOK * sizeof(float), stream);
    scores_head<<<score_grid, gblk, 0, stream>>>(Qr, Qi, Kr, Ki, attn, rowsum, h);
    topk_route<<<N_TOK, 256, 0, stream>>>(attn, rowsum, Vr, Vi, Or, Oi, h);
  }

  cgemm_ow<<<gemm_grid, gblk, 0, stream>>>(Or, Oi, Wo, (float*)d_out);
}